// GNNStack_36928128811712
// MI455X (gfx1250) — compile-verified
//
#include <hip/hip_runtime.h>
#include <math.h>

#define FD 128            // feature dim (D == H == 128)
#define WAVE 32

typedef __attribute__((ext_vector_type(2))) float v2f;
typedef __attribute__((ext_vector_type(8))) float v8f;

// ---------------------------------------------------------------- degrees ---
__global__ void k_deg_init(unsigned* __restrict__ degi, int n) {
    int i = blockIdx.x * blockDim.x + threadIdx.x;
    if (i < n) degi[i] = 1u;                       // self-loop contributes 1
}

__global__ void k_deg_count(const int* __restrict__ dst,
                            unsigned* __restrict__ degi, int e) {
    int i = blockIdx.x * blockDim.x + threadIdx.x;
    if (i < e) atomicAdd(&degi[dst[i]], 1u);
}

__global__ void k_dinv(const unsigned* __restrict__ degi,
                       float* __restrict__ dinv, int n) {
    int i = blockIdx.x * blockDim.x + threadIdx.x;
    if (i < n) dinv[i] = rsqrtf((float)degi[i]);   // deg >= 1 always
}

// ------------------------------------------------------- WMMA f32 GEMM -----
// out[N,128] = A[N,128] @ W[128,128]  (optionally += bias, ReLU)
// grid.x = N/16 row tiles, 256 threads = 8 waves, wave w owns N-tile w.
// V_WMMA_F32_16X16X4_F32: A 16x4 (2 VGPR: lanes0-15 K=0/1, lanes16-31 K=2/3),
// B 4x16 (rows striped across lanes per VGPR), C/D 16x16 in 8 VGPRs.
__global__ __launch_bounds__(256) void k_gemm128_wmma(
    const float* __restrict__ A, const float* __restrict__ W,
    const float* __restrict__ bias, float* __restrict__ out,
    int fuse_bias_relu) {
    const int lane = threadIdx.x & 31;
    const int wid  = threadIdx.x >> 5;       // 0..7 -> 16-col tile
    const int m0   = blockIdx.x << 4;
    const int n0   = wid << 4;
    const int half = lane >> 4;              // 0: lanes 0-15, 1: lanes 16-31
    const int lm   = lane & 15;

    const float* __restrict__ arow = A + (size_t)(m0 + lm) * FD;

    v8f acc0 = {};
    v8f acc1 = {};
#pragma unroll
    for (int k0 = 0; k0 < FD; k0 += 8) {
        // first 16x16x4 slab
        {
            const int ka = k0 + 2 * half;
            v2f a, b;
            a.x = arow[ka];
            a.y = arow[ka + 1];
            b.x = W[(size_t)(ka)     * FD + n0 + lm];
            b.y = W[(size_t)(ka + 1) * FD + n0 + lm];
            acc0 = __builtin_amdgcn_wmma_f32_16x16x4_f32(
                false, a, false, b, (short)0, acc0, false, false);
        }
        // second slab into the other accumulator (breaks WMMA RAW chain)
        {
            const int ka = k0 + 4 + 2 * half;
            v2f a, b;
            a.x = arow[ka];
            a.y = arow[ka + 1];
            b.x = W[(size_t)(ka)     * FD + n0 + lm];
            b.y = W[(size_t)(ka + 1) * FD + n0 + lm];
            acc1 = __builtin_amdgcn_wmma_f32_16x16x4_f32(
                false, a, false, b, (short)0, acc1, false, false);
        }
    }

#pragma unroll
    for (int v = 0; v < 8; ++v) {
        const int row = m0 + v + 8 * half;
        const int col = n0 + lm;
        float r = acc0[v] + acc1[v];
        if (fuse_bias_relu) { r += bias[col]; r = fmaxf(r, 0.0f); }
        out[(size_t)row * FD + col] = r;
    }
}

// ------------------------------------------------ self-loop + bias init ----
// agg[i,f] = hw[i,f] * dinv[i]^2 + bias[f]
__global__ void k_agg_init(const float* __restrict__ hw,
                           const float* __restrict__ dinv,
                           const float* __restrict__ bias,
                           float* __restrict__ agg, int nd) {
    int i = blockIdx.x * blockDim.x + threadIdx.x;
    if (i < nd) {
        const int node = i >> 7;
        const int f    = i & 127;
        const float di = dinv[node];
        agg[i] = hw[i] * di * di + bias[f];
    }
}

// ------------------------------------------------------- edge scatter ------
// one wave per edge: lane loads float4 of hw[src], 4 hw f32 atomic adds.
__global__ __launch_bounds__(256) void k_edge_scatter(
    const int* __restrict__ src, const int* __restrict__ dst,
    const float* __restrict__ dinv, const float* __restrict__ hw,
    float* __restrict__ agg, int e) {
    const int wid  = (int)((blockIdx.x * blockDim.x + threadIdx.x) >> 5);
    const int lane = threadIdx.x & 31;
    if (wid >= e) return;
    const int   s = src[wid];
    const int   d = dst[wid];
    const float w = dinv[s] * dinv[d];
    const float4 v = ((const float4*)(hw + (size_t)s * FD))[lane];
    float* ap = agg + (size_t)d * FD + lane * 4;
    unsafeAtomicAdd(ap + 0, v.x * w);   // global_atomic_add_f32
    unsafeAtomicAdd(ap + 1, v.y * w);
    unsafeAtomicAdd(ap + 2, v.z * w);
    unsafeAtomicAdd(ap + 3, v.w * w);
}

// ------------------------------------------------------------- pointwise ---
__global__ void k_relu(const float* __restrict__ in, float* __restrict__ out,
                       int n) {
    int i = blockIdx.x * blockDim.x + threadIdx.x;
    if (i < n) out[i] = fmaxf(in[i], 0.0f);
}

__global__ void k_emb_relu(const float* __restrict__ agg,
                           float* __restrict__ emb, float* __restrict__ h,
                           int n) {
    int i = blockIdx.x * blockDim.x + threadIdx.x;
    if (i < n) {
        const float v = agg[i];
        emb[i] = v;                  // emb captured before ReLU
        h[i]   = fmaxf(v, 0.0f);
    }
}

// -------------------------------------------- 128 -> 2 head + log_softmax --
// one wave per node, shuffle tree reduction
__global__ __launch_bounds__(256) void k_logits(
    const float* __restrict__ h, const float* __restrict__ Wp2,
    const float* __restrict__ bp2, float* __restrict__ out, int n) {
    const int wid  = (int)((blockIdx.x * blockDim.x + threadIdx.x) >> 5);
    const int lane = threadIdx.x & 31;
    if (wid >= n) return;
    const float4 v = ((const float4*)(h + (size_t)wid * FD))[lane];
    const int f = lane * 4;
    float p0 = v.x * Wp2[(f + 0) * 2 + 0] + v.y * Wp2[(f + 1) * 2 + 0] +
               v.z * Wp2[(f + 2) * 2 + 0] + v.w * Wp2[(f + 3) * 2 + 0];
    float p1 = v.x * Wp2[(f + 0) * 2 + 1] + v.y * Wp2[(f + 1) * 2 + 1] +
               v.z * Wp2[(f + 2) * 2 + 1] + v.w * Wp2[(f + 3) * 2 + 1];
#pragma unroll
    for (int off = 16; off > 0; off >>= 1) {
        p0 += __shfl_down(p0, off, 32);
        p1 += __shfl_down(p1, off, 32);
    }
    if (lane == 0) {
        const float l0 = p0 + bp2[0];
        const float l1 = p1 + bp2[1];
        const float m  = fmaxf(l0, l1);
        const float lse = m + logf(expf(l0 - m) + expf(l1 - m));
        out[(size_t)wid * 2 + 0] = l0 - lse;
        out[(size_t)wid * 2 + 1] = l1 - lse;
    }
}

// ---------------------------------------------------------------------------
extern "C" void kernel_launch(void* const* d_in, const int* in_sizes, int n_in,
                              void* d_out, int out_size, void* d_ws,
                              size_t ws_size, hipStream_t stream) {
    const float* x   = (const float*)d_in[0];
    const int*   ei  = (const int*)d_in[1];
    const float* W1  = (const float*)d_in[2];
    const float* b1  = (const float*)d_in[3];
    const float* W2  = (const float*)d_in[4];
    const float* b2  = (const float*)d_in[5];
    const float* W3  = (const float*)d_in[6];
    const float* b3  = (const float*)d_in[7];
    const float* Wp1 = (const float*)d_in[8];
    const float* bp1 = (const float*)d_in[9];
    const float* Wp2 = (const float*)d_in[10];
    const float* bp2 = (const float*)d_in[11];

    const int N = in_sizes[0] / FD;   // 50000
    const int E = in_sizes[1] / 2;    // 800000
    const int* src = ei;
    const int* dst = ei + E;

    char* ws = (char*)d_ws;
    unsigned* degi = (unsigned*)ws;  ws += (size_t)N * sizeof(unsigned);
    float* dinv = (float*)ws;        ws += (size_t)N * sizeof(float);
    float* hw   = (float*)ws;        ws += (size_t)N * FD * sizeof(float);
    float* agg  = (float*)ws;        ws += (size_t)N * FD * sizeof(float);
    float* h    = (float*)ws;

    float* emb_out  = (float*)d_out;                   // N*128
    float* prob_out = (float*)d_out + (size_t)N * FD;  // N*2

    const dim3 blk(256);
    const int ND      = N * FD;
    const int gN      = (N + 255) / 256;
    const int gE      = (E + 255) / 256;
    const int gND     = (ND + 255) / 256;
    const int gTiles  = N / 16;                 // 3125 (exact)
    const int gEdgeW  = (E * WAVE + 255) / 256; // one wave per edge
    const int gNodeW  = (N * WAVE + 255) / 256; // one wave per node

    // normalization: deg (with self loop) -> d^{-1/2}
    k_deg_init <<<gN, blk, 0, stream>>>(degi, N);
    k_deg_count<<<gE, blk, 0, stream>>>(dst, degi, E);
    k_dinv     <<<gN, blk, 0, stream>>>(degi, dinv, N);

    // layer 1: x -> h
    k_gemm128_wmma<<<gTiles, blk, 0, stream>>>(x, W1, nullptr, hw, 0);
    k_agg_init    <<<gND, blk, 0, stream>>>(hw, dinv, b1, agg, ND);
    k_edge_scatter<<<gEdgeW, blk, 0, stream>>>(src, dst, dinv, hw, agg, E);
    k_relu        <<<gND, blk, 0, stream>>>(agg, h, ND);

    // layer 2: h -> h
    k_gemm128_wmma<<<gTiles, blk, 0, stream>>>(h, W2, nullptr, hw, 0);
    k_agg_init    <<<gND, blk, 0, stream>>>(hw, dinv, b2, agg, ND);
    k_edge_scatter<<<gEdgeW, blk, 0, stream>>>(src, dst, dinv, hw, agg, E);
    k_relu        <<<gND, blk, 0, stream>>>(agg, h, ND);

    // layer 3: h -> emb (pre-ReLU) and h
    k_gemm128_wmma<<<gTiles, blk, 0, stream>>>(h, W3, nullptr, hw, 0);
    k_agg_init    <<<gND, blk, 0, stream>>>(hw, dinv, b3, agg, ND);
    k_edge_scatter<<<gEdgeW, blk, 0, stream>>>(src, dst, dinv, hw, agg, E);
    k_emb_relu    <<<gND, blk, 0, stream>>>(agg, emb_out, h, ND);

    // post-mp: ReLU(h @ Wp1 + bp1) -> hw, then 128->2 + log_softmax
    k_gemm128_wmma<<<gTiles, blk, 0, stream>>>(h, Wp1, bp1, hw, 1);
    k_logits      <<<gNodeW, blk, 0, stream>>>(hw, Wp2, bp2, prob_out, N);
}